// Transformer_20761871909540
// MI455X (gfx1250) — compile-verified
//
#include <hip/hip_runtime.h>
#include <hip/hip_bf16.h>
#include <math.h>

#define Hdim 1024
#define N_GATHER 65536
#define NUM_SEGMENTS 16384

typedef float v2f __attribute__((ext_vector_type(2)));
typedef float v8f __attribute__((ext_vector_type(8)));

// ---------------------------------------------------------------------------
// Kernel 1: gather + ragged segment mean (segment_ids are sorted).
// One block per segment; binary search gives [lo, hi) range of gather indices.
// 256 threads x float4 = 1024 columns per row.
// ---------------------------------------------------------------------------
__device__ __forceinline__ int lower_bound_dev(const int* __restrict__ seg, int n, int val) {
    int lo = 0, hi = n;
    while (lo < hi) {
        int mid = (lo + hi) >> 1;
        if (seg[mid] < val) lo = mid + 1; else hi = mid;
    }
    return lo;
}

__global__ void __launch_bounds__(256) seg_mean_kernel(
    const float* __restrict__ flat,      // [B*S, H]
    const int* __restrict__ gidx,        // [N_GATHER]
    const int* __restrict__ seg,         // [N_GATHER] sorted
    float* __restrict__ hbuf)            // [NUM_SEGMENTS, H]
{
    __shared__ int srange[2];
    const int g = blockIdx.x;
    if (threadIdx.x == 0) {
        srange[0] = lower_bound_dev(seg, N_GATHER, g);
        srange[1] = lower_bound_dev(seg, N_GATHER, g + 1);
    }
    __syncthreads();
    const int lo = srange[0], hi = srange[1];
    const int c = threadIdx.x * 4;

    float4 acc = make_float4(0.f, 0.f, 0.f, 0.f);
    for (int j = lo; j < hi; ++j) {
        const int row = gidx[j];
        const float4 v = *(const float4*)(flat + (size_t)row * Hdim + c);
        acc.x += v.x; acc.y += v.y; acc.z += v.z; acc.w += v.w;
    }
    const float cnt = (float)(hi - lo);
    const float inv = 1.0f / fmaxf(cnt, 1.0f);
    acc.x *= inv; acc.y *= inv; acc.z *= inv; acc.w *= inv;
    *(float4*)(hbuf + (size_t)g * Hdim + c) = acc;
}

// ---------------------------------------------------------------------------
// Kernel 2: z = gelu(h @ W^T + b);  out = z @ proj_w + proj_b
// LDS-blocked f32 WMMA GEMM: BM=64, BN=128, BK=64, fused epilogue.
// Block: 256 threads = 8 waves. Wave w: row-tile rt = w&3 (16 rows),
// col-half cg = (w>>2)*64 (4 col tiles of 16) within the 128-wide N-block.
// ---------------------------------------------------------------------------
#define LDSS 76   // padded row stride in floats: 304B (16B aligned rows,
                  // bank step 12 mod 64 -> 16 distinct banks for frag loads)

__global__ void __launch_bounds__(256, 2) gemm_gelu_proj_kernel(
    const float* __restrict__ hbuf,      // [G, H]
    const float* __restrict__ W,         // [H, H] row-major (dense_w)
    const float* __restrict__ bias,      // [H]
    const float* __restrict__ pw,        // [H]
    const float* __restrict__ pb,        // [1]
    float* __restrict__ out)             // [G]
{
    __shared__ __align__(16) float ldsA[64 * LDSS];
    __shared__ __align__(16) float ldsB[128 * LDSS];
    __shared__ float ldsAcc[64];

    const int tid   = threadIdx.x;
    const int wave  = tid >> 5;
    const int lane  = tid & 31;
    const int l15   = lane & 15;
    const int lhalf = lane >> 4;         // 0: K=k,k+1   1: K=k+2,k+3
    const int rt    = wave & 3;          // row tile within block
    const int cg    = (wave >> 2) * 64;  // column half within N-block
    const int row0  = blockIdx.x * 64;

    if (tid < 64) ldsAcc[tid] = 0.0f;

    float part[8];
#pragma unroll
    for (int r = 0; r < 8; ++r) part[r] = 0.0f;

    for (int nb = 0; nb < 8; ++nb) {          // N-blocks of 128 cols
        const int n0 = nb * 128;
        v8f acc[4];
#pragma unroll
        for (int t = 0; t < 4; ++t)
#pragma unroll
            for (int r = 0; r < 8; ++r) acc[t][r] = 0.0f;

        for (int kb = 0; kb < Hdim / 64; ++kb) {  // K-blocks of 64
            const int k0 = kb * 64;
            __syncthreads();
            // stage A block: 64 rows x 64 K  (16 float4 per row)
#pragma unroll
            for (int i = 0; i < 4; ++i) {
                const int e = tid + i * 256;          // 0..1023
                const int r = e >> 4, c4 = e & 15;
                const float4 v = *(const float4*)(hbuf + (size_t)(row0 + r) * Hdim + k0 + c4 * 4);
                *(float4*)(&ldsA[r * LDSS + c4 * 4]) = v;
            }
            // stage B block: 128 rows of W x 64 K
#pragma unroll
            for (int i = 0; i < 8; ++i) {
                const int e = tid + i * 256;          // 0..2047
                const int r = e >> 4, c4 = e & 15;
                const float4 v = *(const float4*)(W + (size_t)(n0 + r) * Hdim + k0 + c4 * 4);
                *(float4*)(&ldsB[r * LDSS + c4 * 4]) = v;
            }
            __syncthreads();

            // compute: 16 wmma K-steps x 4 column tiles
#pragma unroll
            for (int ks = 0; ks < 16; ++ks) {
                const int kk = ks * 4 + lhalf * 2;
                const v2f a = *(const v2f*)(&ldsA[(rt * 16 + l15) * LDSS + kk]);
#pragma unroll
                for (int t = 0; t < 4; ++t) {
                    const v2f b = *(const v2f*)(&ldsB[(cg + t * 16 + l15) * LDSS + kk]);
                    acc[t] = __builtin_amdgcn_wmma_f32_16x16x4_f32(
                        false, a, false, b, (short)0, acc[t], false, false);
                }
            }
        }

        // fused epilogue for this N-block: bias + exact gelu + proj_w dot
#pragma unroll
        for (int t = 0; t < 4; ++t) {
            const int col = n0 + cg + t * 16 + l15;
            const float bcol = bias[col];
            const float wcol = pw[col];
#pragma unroll
            for (int r = 0; r < 8; ++r) {
                const float x  = acc[t][r] + bcol;
                const float gz = 0.5f * x * (1.0f + erff(x * 0.70710678118654752f));
                part[r] += gz * wcol;
            }
        }
    }

    // reduce across the 16-lane groups (wave32): lanes 0..15 hold rows rt*16+r,
    // lanes 16..31 hold rows rt*16+8+r
#pragma unroll
    for (int off = 1; off < 16; off <<= 1)
#pragma unroll
        for (int r = 0; r < 8; ++r)
            part[r] += __shfl_xor(part[r], off, 32);

    if (l15 == 0) {
#pragma unroll
        for (int r = 0; r < 8; ++r)
            atomicAdd(&ldsAcc[rt * 16 + lhalf * 8 + r], part[r]);
    }
    __syncthreads();
    if (tid < 64) out[row0 + tid] = ldsAcc[tid] + pb[0];
}

// ---------------------------------------------------------------------------
extern "C" void kernel_launch(void* const* d_in, const int* in_sizes, int n_in,
                              void* d_out, int out_size, void* d_ws, size_t ws_size,
                              hipStream_t stream) {
    const float* hid  = (const float*)d_in[0];  // transformer_hidden [B,S,H]
    const float* dW   = (const float*)d_in[1];  // dense_w [H,H]
    const float* db   = (const float*)d_in[2];  // dense_b [H]
    const float* pw   = (const float*)d_in[3];  // proj_w [H]
    const float* pb   = (const float*)d_in[4];  // proj_b [1]
    const int*   gidx = (const int*)d_in[5];    // flat_indices
    const int*   segs = (const int*)d_in[6];    // segment_ids (sorted)
    float* outp = (float*)d_out;
    float* hbuf = (float*)d_ws;                 // [NUM_SEGMENTS, H] = 64 MB

    seg_mean_kernel<<<NUM_SEGMENTS, 256, 0, stream>>>(hid, gidx, segs, hbuf);
    gemm_gelu_proj_kernel<<<NUM_SEGMENTS / 64, 256, 0, stream>>>(hbuf, dW, db, pw, pb, outp);
}